// EfficientSHRFFN_7945689497922
// MI455X (gfx1250) — compile-verified
//
#include <hip/hip_runtime.h>

// Bandwidth-bound row-copy + 4-float patch per 1KB row (B=262144 rows).
// 512 MB total traffic -> ~22us at 23.3 TB/s HBM; compute is negligible, so
// the CDNA5 paths that matter are the async global<->LDS DMA stream
// (ASYNCcnt) and b128/NT cache-control, not WMMA (no contraction exists).

typedef __attribute__((ext_vector_type(4))) float f32x4;
typedef int v4i __attribute__((vector_size(16)));   // matches builtin param type

#define ROWS_PER_BLOCK 4   // 64 threads/row * 4 rows = 256 threads (8 waves, wave32)

#if defined(__gfx1250__) && \
    __has_builtin(__builtin_amdgcn_global_load_async_to_lds_b128) && \
    __has_builtin(__builtin_amdgcn_global_store_async_from_lds_b128)
#define USE_ASYNC_LDS 1
#else
#define USE_ASYNC_LDS 0
#endif

// Pointer-to-AS(1)/AS(3) int4, per the diagnostic:
//   'int __attribute__((vector_size(16))) __device__ *'  (global)
//   expected __shared__ counterpart for the LDS side.
typedef __attribute__((address_space(1))) v4i* gptr_v4i;
typedef __attribute__((address_space(3))) v4i* lptr_v4i;

#define AS_GLOBAL_V4I(p) ((gptr_v4i)(void*)(p))
#define AS_LDS_V4I(p)    ((lptr_v4i)(void*)(p))

__device__ __forceinline__ void wait_asynccnt0() {
#if defined(__gfx1250__)
#if __has_builtin(__builtin_amdgcn_s_wait_asynccnt)
    __builtin_amdgcn_s_wait_asynccnt(0);
#else
    asm volatile("s_wait_asynccnt 0" ::: "memory");
#endif
#endif
}

__global__ __launch_bounds__(256, 8)
void shr_ffn_stream_kernel(const float* __restrict__ x,
                           const int* __restrict__ p_opcode,
                           const int* __restrict__ p_opstart,
                           const int* __restrict__ p_niba,
                           const int* __restrict__ p_nibb,
                           const int* __restrict__ p_result,
                           float* __restrict__ out,
                           int nrows)
{
    // Uniform params: scalar (SMEM) loads, L2-resident.
    const int colA  = *p_niba;
    const int colB  = *p_nibb;
    const int colOp = *p_opstart + *p_opcode;
    const int colR  = *p_result;

    const int tid = threadIdx.x;
    const int r   = tid >> 6;        // row within block (0..3)
    const int l   = tid & 63;        // lane within row  (0..63)
    const int p   = l >> 4;          // pos (0..3)
    const int c4  = l & 15;          // which float4 of the 64-col slice
    const int off = p * 64 + c4 * 4; // element offset within the 256-elem row

    const long long row  = (long long)blockIdx.x * ROWS_PER_BLOCK + r;
    const bool active    = row < (long long)nrows;
    const long long elem = row * 256LL + off;

#if USE_ASYNC_LDS
    // ---- CDNA5 async DMA path: HBM -> LDS -> (patch in LDS) -> HBM ----
    __shared__ float buf[ROWS_PER_BLOCK][256];   // 1KB per row staging

    if (active) {
        // 16B per lane, fully coalesced; data never touches VGPRs.
        __builtin_amdgcn_global_load_async_to_lds_b128(
            AS_GLOBAL_V4I(x + elem), AS_LDS_V4I(&buf[r][off]), 0, 0);
    }
    wait_asynccnt0();      // this wave's LDS writes have landed
    __syncthreads();       // all waves of the block have landed

    // 4 patch threads per row (the ones whose b128 slice covers column RESULT)
    const bool patcher = active && (c4 == (colR >> 2));
    float a0 = 0.f, a1 = 0.f, a2 = 0.f, a3 = 0.f, nb = 0.f, act = 0.f;
    if (patcher) {
        // Read ALL original scalars before any patch write (RESULT may alias).
        a0  = buf[r][colA];
        a1  = buf[r][64 + colA];
        a2  = buf[r][128 + colA];
        a3  = buf[r][192 + colA];
        nb  = buf[r][colB];
        act = buf[r][colOp];
    }
    __syncthreads();

    if (patcher) {
        // value = sum_p a[p]*256^p, mul-then-add in fp32 (match jnp, no FMA)
        float value = __fmul_rn(a0, 1.0f);
        value = __fadd_rn(value, __fmul_rn(a1, 256.0f));
        value = __fadd_rn(value, __fmul_rn(a2, 65536.0f));
        value = __fadd_rn(value, __fmul_rn(a3, 16777216.0f));

        long long iv = (long long)value;            // astype(int64): trunc
        long long sh = (long long)nb;
        sh = sh < 0 ? 0LL : (sh > 31 ? 31LL : sh);  // clip(.,0,31)
        long long shifted = iv >> sh;
        long long ob = (shifted >> (8 * p)) & 255LL;

        buf[r][p * 64 + colR] = __fmul_rn((float)ob, act);  // * op_active
    }
    __syncthreads();

    if (active) {
        __builtin_amdgcn_global_store_async_from_lds_b128(
            AS_GLOBAL_V4I(out + elem), AS_LDS_V4I(&buf[r][off]), 0, 0);
    }
    wait_asynccnt0();      // drain before wave exit (ENDPGM also wait-idles)

#else
    // ---- Fallback: direct VGPR b128 NT streaming ----
    __shared__ float sA[ROWS_PER_BLOCK][4];
    __shared__ float sB[ROWS_PER_BLOCK];
    __shared__ float sOp[ROWS_PER_BLOCK];

    f32x4 v = (f32x4)(0.0f);
    if (active) {
        v = __builtin_nontemporal_load((const f32x4*)(x + elem));
    }

    if (active) {
        if (c4 == (colA >> 2))            sA[r][p] = v[colA & 3];
        if (p == 0 && c4 == (colB >> 2))  sB[r]    = v[colB & 3];
        if (p == 0 && c4 == (colOp >> 2)) sOp[r]   = v[colOp & 3];
    }
    __syncthreads();

    if (active && c4 == (colR >> 2)) {
        float value = __fmul_rn(sA[r][0], 1.0f);
        value = __fadd_rn(value, __fmul_rn(sA[r][1], 256.0f));
        value = __fadd_rn(value, __fmul_rn(sA[r][2], 65536.0f));
        value = __fadd_rn(value, __fmul_rn(sA[r][3], 16777216.0f));

        long long iv = (long long)value;
        long long sh = (long long)sB[r];
        sh = sh < 0 ? 0LL : (sh > 31 ? 31LL : sh);
        long long shifted = iv >> sh;
        long long ob = (shifted >> (8 * p)) & 255LL;

        v[colR & 3] = __fmul_rn((float)ob, sOp[r]);
    }

    if (active) {
        __builtin_nontemporal_store(v, (f32x4*)(out + elem));
    }
#endif
}

extern "C" void kernel_launch(void* const* d_in, const int* in_sizes, int n_in,
                              void* d_out, int out_size, void* d_ws, size_t ws_size,
                              hipStream_t stream) {
    const float* x        = (const float*)d_in[0];
    const int*   opcode   = (const int*)d_in[1];
    const int*   op_start = (const int*)d_in[2];
    const int*   nib_a    = (const int*)d_in[3];
    const int*   nib_b    = (const int*)d_in[4];
    const int*   result   = (const int*)d_in[5];
    float*       out      = (float*)d_out;

    const int nrows  = in_sizes[0] / 256;                 // B (rows of 4x64)
    const int blocks = (nrows + ROWS_PER_BLOCK - 1) / ROWS_PER_BLOCK;

    shr_ffn_stream_kernel<<<blocks, 256, 0, stream>>>(
        x, opcode, op_start, nib_a, nib_b, result, out, nrows);
}